// ExLRestSelfAtten_22557168238928
// MI455X (gfx1250) — compile-verified
//
#include <hip/hip_runtime.h>

#define ATT 5
#define WIN 11
#define HIDDEN 64
#define IN_SIZE 300
#define OUTS 2
#define BB 16
#define SS 2048

typedef __attribute__((ext_vector_type(2))) float v2f;
typedef __attribute__((ext_vector_type(8))) float v8f;
typedef __attribute__((ext_vector_type(4))) unsigned int v4u;
typedef __attribute__((ext_vector_type(4))) int v4i;
typedef __attribute__((ext_vector_type(8))) int v8i;

// D = A(16x4 f32) * B(4x16 f32) + C(16x16 f32), full fp32 WMMA
__device__ __forceinline__ v8f wmma4(v2f a, v2f b, v8f c) {
  return __builtin_amdgcn_wmma_f32_16x16x4_f32(
      /*neg_a=*/false, a, /*neg_b=*/false, b,
      /*c_mod=*/(short)0, c, /*reuse_a=*/false, /*reuse_b=*/false);
}

// Issue one TDM DMA: copy `nelem` contiguous f32 from `gsrc` into LDS at
// byte offset `ldsaddr`.  Built per the CDNA5 D# layout (group0/group1);
// groups 2/3 zero (<=2D tensor).  Tracked by TENSORcnt.
__device__ __forceinline__ void tdm_load_f32(const float* gsrc, unsigned ldsaddr,
                                             unsigned nelem) {
  const unsigned long long ga = (unsigned long long)(const void*)gsrc;
  v4u g0;
  g0[0] = 1u;                                   // count=1, user descriptor
  g0[1] = ldsaddr;                              // LDS byte address
  g0[2] = (unsigned)(ga & 0xffffffffu);         // global_addr[31:0]
  g0[3] = (unsigned)((ga >> 32) & 0x1ffffffu)   // global_addr[56:32]
          | (2u << 30);                         // type = 2 ("image")
  v8i g1 = {};
  g1[0] = (int)(2u << 16);                      // workgroup_mask=0, data_size=4B
  g1[1] = (int)((nelem & 0xffffu) << 16);       // tensor_dim0 lo16
  g1[2] = (int)(((nelem >> 16) & 0xffffu)       // tensor_dim0 hi16
          | (1u << 16));                        // tensor_dim1 = 1
  g1[3] = (int)(nelem << 16);                   // tile_dim0 = nelem (1-D tile)
  g1[5] = (int)nelem;                           // tensor_dim0_stride
  const v4i z4 = {};
#if defined(__clang_major__) && (__clang_major__ >= 23)
  const v8i z8 = {};
  __builtin_amdgcn_tensor_load_to_lds(g0, g1, z4, z4, z8, 0);
#else
  __builtin_amdgcn_tensor_load_to_lds(g0, g1, z4, z4, 0);
#endif
}

// -------------------------------------------------------------------------
// Kernel 1: h = relu(x @ W1 + b1)    rows = B*S, K = 300, N = 64
// -------------------------------------------------------------------------
__global__ __launch_bounds__(32) void k_hidden(const float* __restrict__ x,
                                               const float* __restrict__ W1,
                                               const float* __restrict__ b1,
                                               float* __restrict__ h) {
  const int lane = threadIdx.x;
  const int half = lane >> 4, l16 = lane & 15;
  const int row0 = blockIdx.x * 16;  // global row over B*S
  v8f a0 = {}, a1 = {}, a2 = {}, a3 = {};
  const float* xr = x + (size_t)(row0 + l16) * IN_SIZE;
  for (int k0 = 0; k0 < IN_SIZE; k0 += 4) {  // 300 = 75 * 4
    const int ka = k0 + 2 * half;
    v2f a; a.x = xr[ka]; a.y = xr[ka + 1];
    const float* w0 = W1 + (size_t)ka * HIDDEN + l16;
    const float* w1 = w0 + HIDDEN;
    v2f b;
    b.x = w0[0];  b.y = w1[0];  a0 = wmma4(a, b, a0);
    b.x = w0[16]; b.y = w1[16]; a1 = wmma4(a, b, a1);
    b.x = w0[32]; b.y = w1[32]; a2 = wmma4(a, b, a2);
    b.x = w0[48]; b.y = w1[48]; a3 = wmma4(a, b, a3);
  }
#pragma unroll
  for (int i = 0; i < 8; i++) {
    float* hr = h + (size_t)(row0 + i + 8 * half) * HIDDEN + l16;
    hr[0]  = fmaxf(a0[i] + b1[l16],      0.f);
    hr[16] = fmaxf(a1[i] + b1[16 + l16], 0.f);
    hr[32] = fmaxf(a2[i] + b1[32 + l16], 0.f);
    hr[48] = fmaxf(a3[i] + b1[48 + l16], 0.f);
  }
}

// -------------------------------------------------------------------------
// Kernel 2: n = 11-tap box-sum of h (per batch, zero padded);
//           q = h@Wq ; K = n@Wk ; V = n@Wv
// -------------------------------------------------------------------------
__global__ __launch_bounds__(32) void k_qkv(const float* __restrict__ h,
                                            const float* __restrict__ Wq,
                                            const float* __restrict__ Wk,
                                            const float* __restrict__ Wv,
                                            float* __restrict__ q,
                                            float* __restrict__ kmat,
                                            float* __restrict__ vmat) {
  __shared__ float hext[(16 + 2 * ATT) * HIDDEN];  // rows s0-5 .. s0+20
  __shared__ float nloc[16 * HIDDEN];
  const int lane = threadIdx.x, half = lane >> 4, l16 = lane & 15;
  const int b = blockIdx.x >> 7;             // 128 tiles per batch
  const int s0 = (blockIdx.x & 127) * 16;

  for (int idx = lane; idx < (16 + 2 * ATT) * HIDDEN; idx += 32) {
    const int j = idx >> 6, c = idx & 63;
    const int s = s0 - ATT + j;
    hext[idx] = (s >= 0 && s < SS) ? h[((size_t)b * SS + s) * HIDDEN + c] : 0.f;
  }
  __syncthreads();
  for (int idx = lane; idx < 16 * HIDDEN; idx += 32) {
    const int r = idx >> 6, c = idx & 63;
    float acc = 0.f;
#pragma unroll
    for (int j = 0; j < WIN; j++) acc += hext[(r + j) * HIDDEN + c];
    nloc[idx] = acc;
  }
  __syncthreads();

  auto gemm16 = [&](const float* As, const float* Wg, float* Og) {
    v8f a0 = {}, a1 = {}, a2 = {}, a3 = {};
    for (int k0 = 0; k0 < HIDDEN; k0 += 4) {
      const int ka = k0 + 2 * half;
      v2f a; a.x = As[l16 * HIDDEN + ka]; a.y = As[l16 * HIDDEN + ka + 1];
      const float* w0 = Wg + (size_t)ka * HIDDEN + l16;
      const float* w1 = w0 + HIDDEN;
      v2f bb;
      bb.x = w0[0];  bb.y = w1[0];  a0 = wmma4(a, bb, a0);
      bb.x = w0[16]; bb.y = w1[16]; a1 = wmma4(a, bb, a1);
      bb.x = w0[32]; bb.y = w1[32]; a2 = wmma4(a, bb, a2);
      bb.x = w0[48]; bb.y = w1[48]; a3 = wmma4(a, bb, a3);
    }
#pragma unroll
    for (int i = 0; i < 8; i++) {
      float* o = Og + (size_t)(i + 8 * half) * HIDDEN + l16;
      o[0] = a0[i]; o[16] = a1[i]; o[32] = a2[i]; o[48] = a3[i];
    }
  };
  const size_t base = ((size_t)b * SS + s0) * HIDDEN;
  gemm16(hext + ATT * HIDDEN, Wq, q + base);
  gemm16(nloc, Wk, kmat + base);
  gemm16(nloc, Wv, vmat + base);
}

// -------------------------------------------------------------------------
// Kernel 3: raw scores S[s,t] = q[s,:]·K[t,:] per batch.
// 4 waves / block, each owning a 16-row s-tile; K-tiles staged into LDS by
// the Tensor Data Mover with double-buffered prefetch (TENSORcnt pipeline):
// wave 0 issues the DMA for tile t+1 into the alternate buffer, then waits
// tensorcnt<=1 (in-order completion) before the block consumes tile t.
// Raw tiles go straight to the atten output region; FlashAttention-style
// online (rowmax m, sumexp l) per s-row.
// -------------------------------------------------------------------------
__global__ __launch_bounds__(128) void k_scores(const float* __restrict__ q,
                                                const float* __restrict__ kmat,
                                                float* __restrict__ atten,
                                                float* __restrict__ mrow,
                                                float* __restrict__ lrow) {
  __shared__ float kt[2][16 * HIDDEN];  // double-buffered 16x64 K tiles
  const int tid = threadIdx.x;
  const int warp = tid >> 5, lane = tid & 31;
  const int half = lane >> 4, l16 = lane & 15;
  const int b = blockIdx.x >> 5;                  // 32 blocks per batch
  const int s0 = (blockIdx.x & 31) * 64 + warp * 16;

  // cache the full 16x64 q tile as prebuilt A operands (32 VGPRs)
  v2f qa[16];
  const float* qr = q + ((size_t)b * SS + s0 + l16) * HIDDEN;
#pragma unroll
  for (int kk = 0; kk < 16; kk++) {
    const int ka = kk * 4 + 2 * half;
    qa[kk].x = qr[ka]; qa[kk].y = qr[ka + 1];
  }
  float m8[8], l8[8];
#pragma unroll
  for (int i = 0; i < 8; i++) { m8[i] = -__builtin_inff(); l8[i] = 0.f; }

  float* arow = atten + ((size_t)b * SS + s0) * SS;
  const float* kbat = kmat + (size_t)b * SS * HIDDEN;
  // LDS aperture: low 32 address bits are the LDS byte offset
  const unsigned ldsA = (unsigned)(size_t)(void*)&kt[0][0];
  const unsigned ldsB = (unsigned)(size_t)(void*)&kt[1][0];

  if (warp == 0) tdm_load_f32(kbat, ldsA, 16 * HIDDEN);  // prefetch tile 0

  for (int tt = 0; tt < SS / 16; tt++) {
    const int t0 = tt * 16;
    if (warp == 0) {
      if (tt + 1 < SS / 16) {
        // prefetch next tile into the other buffer, then wait for current
        tdm_load_f32(kbat + (size_t)(t0 + 16) * HIDDEN,
                     (tt & 1) ? ldsA : ldsB, 16 * HIDDEN);
        __builtin_amdgcn_s_wait_tensorcnt(1);
      } else {
        __builtin_amdgcn_s_wait_tensorcnt(0);
      }
    }
    __syncthreads();                       // tile tt visible to all waves
    const float* kb = kt[tt & 1];
    v8f acc = {};
#pragma unroll
    for (int kk = 0; kk < 16; kk++) {      // B = K^T : B[d][t] = K[t][d]
      const int ka = kk * 4 + 2 * half;
      v2f bb; bb.x = kb[l16 * HIDDEN + ka]; bb.y = kb[l16 * HIDDEN + ka + 1];
      acc = wmma4(qa[kk], bb, acc);
    }
#pragma unroll
    for (int i = 0; i < 8; i++) {          // per-lane online softmax stats
      const float sv = acc[i];
      arow[(size_t)(i + 8 * half) * SS + t0 + l16] = sv;  // raw score
      const float mn = fmaxf(m8[i], sv);
      l8[i] = l8[i] * __expf(m8[i] - mn) + __expf(sv - mn);
      m8[i] = mn;
    }
    __syncthreads();  // all ds reads of this buffer retired before reuse
  }
  // merge (m,l) across the 16 lanes of each half-wave
#pragma unroll
  for (int i = 0; i < 8; i++) {
    float m = m8[i], l = l8[i];
#pragma unroll
    for (int off = 1; off < 16; off <<= 1) {
      const float mo = __shfl_xor(m, off, 32);
      const float lo = __shfl_xor(l, off, 32);
      const float mn = fmaxf(m, mo);
      l = l * __expf(m - mn) + lo * __expf(mo - mn);
      m = mn;
    }
    m8[i] = m; l8[i] = l;
  }
  if (l16 == 0) {
#pragma unroll
    for (int i = 0; i < 8; i++) {
      mrow[(size_t)b * SS + s0 + 8 * half + i] = m8[i];
      lrow[(size_t)b * SS + s0 + 8 * half + i] = l8[i];
    }
  }
}

// -------------------------------------------------------------------------
// Kernel 4: per (b, 16-wide t-tile): normalize A in place
//           (A = exp(S-m)/l/8), accumulate AV = A^T @ V via WMMA (K = s),
//           then out = AV @ W2 + b2 via cross-lane reduction.
// -------------------------------------------------------------------------
__global__ __launch_bounds__(32) void k_out(const float* __restrict__ vmat,
                                            const float* __restrict__ mrow,
                                            const float* __restrict__ lrow,
                                            const float* __restrict__ W2,
                                            const float* __restrict__ b2,
                                            float* __restrict__ atten,
                                            float* __restrict__ out) {
  const int lane = threadIdx.x, half = lane >> 4, l16 = lane & 15;
  const int b = blockIdx.x >> 7;
  const int t0 = (blockIdx.x & 127) * 16;
  v8f a0 = {}, a1 = {}, a2 = {}, a3 = {};
  const float* mr = mrow + (size_t)b * SS;
  const float* lr = lrow + (size_t)b * SS;
  float* abase = atten + (size_t)b * SS * SS + t0;     // column block [.., t0+15]
  const float* vb = vmat + (size_t)b * SS * HIDDEN;

  for (int sc = 0; sc < SS; sc += 4) {                 // K dimension = s
    const int sa = sc + 2 * half;
    const float r0 = abase[(size_t)sa * SS + l16];
    const float r1 = abase[(size_t)(sa + 1) * SS + l16];
    const float n0 = __expf(r0 - mr[sa])     / (lr[sa]     * 8.0f);
    const float n1 = __expf(r1 - mr[sa + 1]) / (lr[sa + 1] * 8.0f);
    abase[(size_t)sa * SS + l16] = n0;                 // final softmax/8 output
    abase[(size_t)(sa + 1) * SS + l16] = n1;
    v2f a; a.x = n0; a.y = n1;                         // A^T tile: M=t, K=s
    const float* v0 = vb + (size_t)sa * HIDDEN + l16;
    const float* v1 = v0 + HIDDEN;
    v2f bb;
    bb.x = v0[0];  bb.y = v1[0];  a0 = wmma4(a, bb, a0);
    bb.x = v0[16]; bb.y = v1[16]; a1 = wmma4(a, bb, a1);
    bb.x = v0[32]; bb.y = v1[32]; a2 = wmma4(a, bb, a2);
    bb.x = v0[48]; bb.y = v1[48]; a3 = wmma4(a, bb, a3);
  }
  // AV(16 t-rows x 64) -> out2 = AV @ W2 + b2 (64x2)
  const float w0a = W2[l16 * 2],        w0b = W2[l16 * 2 + 1];
  const float w1a = W2[(16 + l16) * 2], w1b = W2[(16 + l16) * 2 + 1];
  const float w2a = W2[(32 + l16) * 2], w2b = W2[(32 + l16) * 2 + 1];
  const float w3a = W2[(48 + l16) * 2], w3b = W2[(48 + l16) * 2 + 1];
#pragma unroll
  for (int i = 0; i < 8; i++) {
    float p0 = a0[i] * w0a + a1[i] * w1a + a2[i] * w2a + a3[i] * w3a;
    float p1 = a0[i] * w0b + a1[i] * w1b + a2[i] * w2b + a3[i] * w3b;
#pragma unroll
    for (int off = 1; off < 16; off <<= 1) {
      p0 += __shfl_xor(p0, off, 32);
      p1 += __shfl_xor(p1, off, 32);
    }
    if (l16 == 0) {
      const int t = t0 + 8 * half + i;
      out[((size_t)b * SS + t) * OUTS + 0] = p0 + b2[0];
      out[((size_t)b * SS + t) * OUTS + 1] = p1 + b2[1];
    }
  }
}

// -------------------------------------------------------------------------
extern "C" void kernel_launch(void* const* d_in, const int* in_sizes, int n_in,
                              void* d_out, int out_size, void* d_ws, size_t ws_size,
                              hipStream_t stream) {
  (void)in_sizes; (void)n_in; (void)out_size; (void)ws_size;
  const float* x  = (const float*)d_in[0];
  const float* W1 = (const float*)d_in[1];
  const float* b1 = (const float*)d_in[2];
  const float* W2 = (const float*)d_in[3];
  const float* b2 = (const float*)d_in[4];
  const float* Wq = (const float*)d_in[5];
  const float* Wk = (const float*)d_in[6];
  const float* Wv = (const float*)d_in[7];

  float* out   = (float*)d_out;                       // (B,S,2)
  float* atten = out + (size_t)BB * SS * OUTS;        // (B,S,S)

  float* ws   = (float*)d_ws;                         // ~33.8 MB used
  float* h    = ws;
  float* q    = h    + (size_t)BB * SS * HIDDEN;
  float* kmat = q    + (size_t)BB * SS * HIDDEN;
  float* vmat = kmat + (size_t)BB * SS * HIDDEN;
  float* mrow = vmat + (size_t)BB * SS * HIDDEN;
  float* lrow = mrow + (size_t)BB * SS;

  const int nTiles = BB * SS / 16;  // 2048 16-row tiles
  k_hidden<<<nTiles, dim3(32), 0, stream>>>(x, W1, b1, h);
  k_qkv   <<<nTiles, dim3(32), 0, stream>>>(h, Wq, Wk, Wv, q, kmat, vmat);
  k_scores<<<BB * SS / 64, dim3(128), 0, stream>>>(q, kmat, atten, mrow, lrow);
  k_out   <<<nTiles, dim3(32), 0, stream>>>(vmat, mrow, lrow, W2, b2, atten, out);
}